// NTMReadController_70574902608301
// MI455X (gfx1250) — compile-verified
//
#include <hip/hip_runtime.h>
#include <hip/hip_bf16.h>
#include <math.h>

#define EPSV 1e-16f
#define N_DIM 1024
#define M_DIM 64
#define MSTRIDE 65      // pad rows to 65 floats -> bank-conflict-free on 64 banks
#define E_DIM 512
#define P_DIM 70
#define P_STRIDE 80
#define BLOCK 256
#define RPT 4           // N_DIM / BLOCK

typedef __attribute__((ext_vector_type(2))) float v2f;
typedef __attribute__((ext_vector_type(8))) float v8f;

// ---------------------------------------------------------------------------
// Kernel 1: proj = embeddings(1024x512) @ W(512x70) + b, via V_WMMA_F32_16X16X4_F32
// One wave per 16x16 output tile. N padded 70 -> 80 (5 tiles of 16).
// A layout (16x4 f32): lanes 0-15 hold K={k0,k0+1}, lanes 16-31 hold K={k0+2,k0+3}, M=lane%16.
// B layout (4x16 f32): lanes 0-15 rows K={k0,k0+1}, lanes 16-31 rows K={k0+2,k0+3}, N=lane%16.
// C/D (16x16 f32, 8 VGPRs): M = vgpr + 8*(lane>=16), N = lane%16.
// Padding lanes (col>=70) load a clamped in-range W column and select 0 —
// fully branchless so EXEC stays all-1s through the whole loop (WMMA req).
// ---------------------------------------------------------------------------
__global__ __launch_bounds__(32) void proj_wmma_kernel(
    const float* __restrict__ emb, const float* __restrict__ W,
    const float* __restrict__ bias, float* __restrict__ proj)
{
    const int lane  = threadIdx.x;        // 0..31
    const int mtile = blockIdx.x;         // 0..63
    const int ntile = blockIdx.y;         // 0..4
    const int l15   = lane & 15;
    const int khalf = (lane >> 4) << 1;   // 0 or 2
    const int arow  = mtile * 16 + l15;   // A row for this lane
    const int col   = ntile * 16 + l15;   // output column
    const bool colv = (col < P_DIM);
    const int  colc = colv ? col : 0;     // clamped, always in range
    const float msk = colv ? 1.0f : 0.0f;

    const float* __restrict__ arowp = emb + (size_t)arow * E_DIM + khalf;
    const float* __restrict__ wcolp = W + colc;

    v8f acc = {};
#if __has_builtin(__builtin_amdgcn_wmma_f32_16x16x4_f32)
    for (int k0 = 0; k0 < E_DIM; k0 += 4) {
        const int ka = k0 + khalf;
        // A: 8B-aligned pair load (index arow*512 + ka is even)
        const float2 av = *(const float2*)(arowp + k0);
        v2f a, bv;
        a.x = av.x;
        a.y = av.y;
        // B: unconditional in-range loads, masked by select (no EXEC divergence)
        bv.x = wcolp[(size_t)ka * P_DIM] * msk;
        bv.y = wcolp[(size_t)(ka + 1) * P_DIM] * msk;
        // (neg_a, A, neg_b, B, c_mod, C, reuse_a, reuse_b)
        acc = __builtin_amdgcn_wmma_f32_16x16x4_f32(
            false, a, false, bv, (short)0, acc, false, false);
    }
#else
    // Scalar fallback: each lane computes its 8 C/D elements directly.
    if (colv) {
        const int rbase = mtile * 16 + ((lane >> 4) << 3);
        for (int k = 0; k < E_DIM; ++k) {
            const float wv = W[k * P_DIM + col];
            #pragma unroll
            for (int i = 0; i < 8; ++i)
                acc[i] = fmaf(emb[(rbase + i) * E_DIM + k], wv, acc[i]);
        }
    }
#endif
    if (colv) {
        const float bb = bias[col];
        const int rbase = mtile * 16 + ((lane >> 4) << 3);
        #pragma unroll
        for (int i = 0; i < 8; ++i)
            proj[(size_t)(rbase + i) * P_STRIDE + col] = acc[i] + bb;
    }
}

// ---------------------------------------------------------------------------
// Block reductions through LDS scratch (RED has BLOCK slots)
// ---------------------------------------------------------------------------
__device__ __forceinline__ float block_reduce_sum(float v, float* RED, int tid) {
    RED[tid] = v; __syncthreads();
    for (int s = BLOCK / 2; s > 0; s >>= 1) {
        if (tid < s) RED[tid] += RED[tid + s];
        __syncthreads();
    }
    float r = RED[0]; __syncthreads();
    return r;
}
__device__ __forceinline__ float block_reduce_max(float v, float* RED, int tid) {
    RED[tid] = v; __syncthreads();
    for (int s = BLOCK / 2; s > 0; s >>= 1) {
        if (tid < s) RED[tid] = fmaxf(RED[tid], RED[tid + s]);
        __syncthreads();
    }
    float r = RED[0]; __syncthreads();
    return r;
}

__device__ __forceinline__ float softplusf(float x) {
    return (x > 30.0f) ? x : log1pf(expf(x));
}

// ---------------------------------------------------------------------------
// Kernel 2: fused NTM addressing + read. One workgroup per batch row.
// memory[b] (1024x64 f32 = 256KB) staged once in LDS (stride-65 padded).
// HBM traffic: memory read exactly once -> ~11.5us floor at 23.3 TB/s.
// ---------------------------------------------------------------------------
__global__ __launch_bounds__(BLOCK) void ntm_fused_kernel(
    const float* __restrict__ w_prev, const float* __restrict__ memory,
    const float* __restrict__ proj, float* __restrict__ out_data,
    float* __restrict__ out_w)
{
    extern __shared__ float lds[];
    float* MEM = lds;                          // N_DIM * MSTRIDE
    float* KV  = MEM + N_DIM * MSTRIDE;        // 64
    float* WG  = KV + M_DIM;                   // N_DIM
    float* RED = WG + N_DIM;                   // BLOCK
    float* SC  = RED + BLOCK;                  // 8 scalars

    const int tid = threadIdx.x;
    const int b   = blockIdx.x;
    const float* projrow = proj + (size_t)b * P_STRIDE;
    const float* memrow  = memory + (size_t)b * (N_DIM * M_DIM);

    // --- gate/scalar head (thread 0) + key vector ---
    if (tid == 0) {
        const float pb = projrow[64], pg = projrow[65];
        const float s0 = projrow[66], s1 = projrow[67], s2 = projrow[68];
        const float py = projrow[69];
        const float sm = fmaxf(s0, fmaxf(s1, s2));
        const float e0 = expf(s0 - sm), e1 = expf(s1 - sm), e2 = expf(s2 - sm);
        const float es = e0 + e1 + e2;
        SC[0] = softplusf(pb);                 // beta
        SC[1] = 1.0f / (1.0f + expf(-pg));     // g
        SC[2] = e0 / es; SC[3] = e1 / es; SC[4] = e2 / es;   // s
        SC[5] = 1.0f + softplusf(py);          // y
    }
    if (tid < M_DIM) KV[tid] = projrow[tid];

    // --- stage memory[b] into LDS (float4 global loads, padded rows) ---
    for (int i = tid; i < N_DIM * M_DIM / 4; i += BLOCK) {
        const int row = i >> 4;
        const int m4  = (i & 15) << 2;
        const float4 v = *(const float4*)(memrow + row * M_DIM + m4);
        float* d = MEM + row * MSTRIDE + m4;
        d[0] = v.x; d[1] = v.y; d[2] = v.z; d[3] = v.w;
    }
    __syncthreads();

    const float beta = SC[0], g = SC[1];
    const float s0 = SC[2], s1 = SC[3], s2 = SC[4], y = SC[5];

    // k_norm (broadcast LDS reads, redundant per-thread; trivial cost)
    float ks = 0.0f;
    for (int m = 0; m < M_DIM; ++m) { const float kv = KV[m]; ks = fmaf(kv, kv, ks); }
    const float knorm = sqrtf(ks);

    // --- cosine similarity, z = beta * sim ---
    float zloc[RPT];
    float zmaxl = -INFINITY;
    #pragma unroll
    for (int r = 0; r < RPT; ++r) {
        const int n = tid + r * BLOCK;
        const float* mr = MEM + n * MSTRIDE;
        float dot = 0.0f, nrm = 0.0f;
        for (int m = 0; m < M_DIM; ++m) {
            const float mv = mr[m];
            dot = fmaf(KV[m], mv, dot);
            nrm = fmaf(mv, mv, nrm);
        }
        const float sim = dot / (knorm * sqrtf(nrm) + EPSV);
        zloc[r] = beta * sim;
        zmaxl = fmaxf(zmaxl, zloc[r]);
    }
    const float zmax = block_reduce_max(zmaxl, RED, tid);

    // --- softmax over n, then gate-interpolate with w_prev ---
    float el[RPT], esuml = 0.0f;
    #pragma unroll
    for (int r = 0; r < RPT; ++r) { el[r] = expf(zloc[r] - zmax); esuml += el[r]; }
    const float esum = block_reduce_sum(esuml, RED, tid);
    #pragma unroll
    for (int r = 0; r < RPT; ++r) {
        const int n = tid + r * BLOCK;
        const float wc = el[r] / esum;
        WG[n] = g * wc + (1.0f - g) * w_prev[(size_t)b * N_DIM + n];
    }
    __syncthreads();

    // --- circular shift (roll -1 / 0 / +1) ---
    float wt[RPT];
    #pragma unroll
    for (int r = 0; r < RPT; ++r) {
        const int n  = tid + r * BLOCK;
        const int np = (n + 1) & (N_DIM - 1);
        const int nm = (n + N_DIM - 1) & (N_DIM - 1);
        wt[r] = s0 * WG[np] + s1 * WG[n] + s2 * WG[nm];
    }
    __syncthreads();

    // --- sharpen: w = wt^y / (sum + eps) ---
    float wp[RPT], psuml = 0.0f;
    #pragma unroll
    for (int r = 0; r < RPT; ++r) { wp[r] = powf(wt[r], y); psuml += wp[r]; }
    const float psum = block_reduce_sum(psuml, RED, tid);
    const float inv = 1.0f / (psum + EPSV);
    #pragma unroll
    for (int r = 0; r < RPT; ++r) {
        const int n = tid + r * BLOCK;
        const float w = wp[r] * inv;
        WG[n] = w;
        out_w[(size_t)b * N_DIM + n] = w;
    }
    __syncthreads();

    // --- weighted read: out_data[b,m] = sum_n w[n] * mem[n,m] ---
    // 4 groups of 64 threads; group gid covers n in [gid*256, gid*256+256).
    const int gid = tid >> 6;     // 0..3
    const int m   = tid & 63;     // column; stride-65 rows -> conflict-free
    float part = 0.0f;
    const int n0 = gid * (N_DIM / 4);
    for (int n = n0; n < n0 + N_DIM / 4; ++n)
        part = fmaf(WG[n], MEM[n * MSTRIDE + m], part);
    RED[gid * 64 + m] = part;
    __syncthreads();
    if (tid < M_DIM) {
        out_data[(size_t)b * M_DIM + tid] =
            RED[tid] + RED[64 + tid] + RED[128 + tid] + RED[192 + tid];
    }
}

// ---------------------------------------------------------------------------
extern "C" void kernel_launch(void* const* d_in, const int* in_sizes, int n_in,
                              void* d_out, int out_size, void* d_ws, size_t ws_size,
                              hipStream_t stream) {
    const float* emb    = (const float*)d_in[0];   // (1024, 512)
    const float* w_prev = (const float*)d_in[1];   // (1024, 1024)
    const float* memory = (const float*)d_in[2];   // (1024, 1024, 64)
    const float* W      = (const float*)d_in[3];   // (512, 70)
    const float* bias   = (const float*)d_in[4];   // (70,)

    float* proj     = (float*)d_ws;                // (1024, 80) padded
    float* out_data = (float*)d_out;               // (1024, 64)
    float* out_w    = (float*)d_out + 1024 * 64;   // (1024, 1024)

    dim3 g1(64, 5, 1);
    proj_wmma_kernel<<<g1, 32, 0, stream>>>(emb, W, bias, proj);

    const size_t ldsBytes =
        (size_t)(N_DIM * MSTRIDE + M_DIM + N_DIM + BLOCK + 8) * sizeof(float);
    ntm_fused_kernel<<<1024, BLOCK, ldsBytes, stream>>>(
        w_prev, memory, proj, out_data, out_w);
}